// MSDeformAttn_9371618640483
// MI455X (gfx1250) — compile-verified
//
#include <hip/hip_runtime.h>
#include <hip/hip_bf16.h>

// MSDeformAttn for MI455X (gfx1250, wave32).
// Pipeline: 3x fp32 WMMA GEMMs -> per-(n,q,h) softmax+bilinear sampling -> 1x WMMA GEMM.

typedef __attribute__((ext_vector_type(2))) float v2f;
typedef __attribute__((ext_vector_type(8))) float v8f;

#define MSDA_N    4
#define MSDA_LQ   5440
#define MSDA_C    256
#define MSDA_H    8
#define MSDA_L    4
#define MSDA_P    4
#define MSDA_D    32
#define MSDA_M    (MSDA_N * MSDA_LQ)   // 21760 rows (LQ == LEN_IN here)

// ---------------------------------------------------------------------------
// Generic C = A(MxK) * B(KxN) + bias, fp32, K fixed at 256.
// Block: 256 threads (8 waves). Tile: 128 rows x 64 cols.
// Each wave: 16 rows x 64 cols via 4 accumulators of V_WMMA_F32_16X16X4_F32.
// LDS B panel is stored pair-interleaved: lds2[k/2][n] = (B[k][n], B[k+1][n])
// so each per-lane B fragment is a single naturally-aligned ds_load_b64 that
// lands in an even-aligned VGPR pair (WMMA operand requirement) with no moves.
// ---------------------------------------------------------------------------
__global__ __launch_bounds__(256) void gemm_f32_wmma(
    const float* __restrict__ A, const float* __restrict__ B,
    const float* __restrict__ bias, float* __restrict__ C,
    int Nout)
{
    __shared__ v2f ldsB2[128 * 64];   // 64 KB: (K/2) pair-rows x 64 cols

    const int tid   = threadIdx.x;
    const int mBase = blockIdx.x * 128;
    const int nBase = blockIdx.y * 64;

    // Cooperative fill: each thread handles two consecutive B rows (a k-pair)
    // for 4 columns -> 2 coalesced float4 loads + 4 ds_store_b64.
    {
        const int pr = tid >> 4;          // pair-row within pass, 0..15
        const int q  = (tid & 15) << 2;   // 0,4,...,60
        #pragma unroll
        for (int pass = 0; pass < 8; ++pass) {
            const int k2 = pass * 16 + pr;            // 0..127
            const float4 r0 = *(const float4*)(B + (size_t)(2 * k2)     * Nout + nBase + q);
            const float4 r1 = *(const float4*)(B + (size_t)(2 * k2 + 1) * Nout + nBase + q);
            v2f* dst = &ldsB2[k2 * 64 + q];
            dst[0] = (v2f){r0.x, r1.x};
            dst[1] = (v2f){r0.y, r1.y};
            dst[2] = (v2f){r0.z, r1.z};
            dst[3] = (v2f){r0.w, r1.w};
        }
    }
    __syncthreads();

    const int wid   = tid >> 5;
    const int lane  = tid & 31;
    const int lhalf = lane >> 4;    // 0: K={0,1}; 1: K={2,3} within a k-step
    const int l15   = lane & 15;

    v8f acc[4];
    #pragma unroll
    for (int t = 0; t < 4; ++t) {
        const float bv = bias[nBase + t * 16 + l15];
        #pragma unroll
        for (int v = 0; v < 8; ++v) acc[t][v] = bv;
    }

    const int aRow = mBase + wid * 16 + l15;
    const float* aPtr = A + (size_t)aRow * 256 + (lhalf << 1);

    #pragma unroll 4
    for (int k = 0; k < 256; k += 4) {
        const int k2 = (k >> 1) + lhalf;             // pair-row index for this lane
        const v2f a = *(const v2f*)(aPtr + k);       // A[row][kk], A[row][kk+1]
        #pragma unroll
        for (int t = 0; t < 4; ++t) {
            const v2f b = ldsB2[k2 * 64 + t * 16 + l15];   // (B[kk][col], B[kk+1][col])
            acc[t] = __builtin_amdgcn_wmma_f32_16x16x4_f32(
                false, a, false, b, (short)0, acc[t], false, false);
        }
    }

    #pragma unroll
    for (int t = 0; t < 4; ++t) {
        const int col = nBase + t * 16 + l15;
        #pragma unroll
        for (int v = 0; v < 8; ++v) {
            const int row = mBase + wid * 16 + (lhalf << 3) + v;
            C[(size_t)row * Nout + col] = acc[t][v];
        }
    }
}

// ---------------------------------------------------------------------------
// Softmax over 16 points + bilinear sampling. One wave32 per (n, q, h);
// the 32 lanes are the D=32 channels -> each corner gather is a coalesced
// 128B read (L2-resident: value tensor is 22MB vs 192MB L2).
// ---------------------------------------------------------------------------
__global__ __launch_bounds__(256) void msda_sample(
    const float* __restrict__ ref,    // (N, LQ, L, 2)
    const float* __restrict__ offb,   // (M, 256) = (n,q, h*L*P*2)
    const float* __restrict__ attnb,  // (M, 128) = (n,q, h*16)
    const float* __restrict__ value,  // (M, 256) = (n,pos, h*32+d)
    float*       __restrict__ inter,  // (M, 256)
    const int*   __restrict__ shapes, // (L,2)
    const int*   __restrict__ starts) // (L,)
{
    const int gtid = blockIdx.x * blockDim.x + threadIdx.x;
    const int wave = gtid >> 5;
    const int lane = gtid & 31;                 // channel d
    const int h    = wave & (MSDA_H - 1);
    const int q    = (wave >> 3) % MSDA_LQ;
    const int n    = wave / (MSDA_H * MSDA_LQ);
    const int row  = n * MSDA_LQ + q;

    // Softmax over the 16 attention logits (uniform per wave).
    const float* al = attnb + (size_t)row * 128 + h * 16;
    float e[16];
    float m = al[0];
    #pragma unroll
    for (int i = 1; i < 16; ++i) m = fmaxf(m, al[i]);
    float s = 0.0f;
    #pragma unroll
    for (int i = 0; i < 16; ++i) { e[i] = __expf(al[i] - m); s += e[i]; }
    const float inv = 1.0f / s;

    const float* off = offb + (size_t)row * 256 + h * 32;
    float acc = 0.0f;

    #pragma unroll
    for (int l = 0; l < MSDA_L; ++l) {
        const int hh = shapes[2 * l];
        const int ww = shapes[2 * l + 1];
        const int s0 = starts[l];
        const float rx = ref[((size_t)row * MSDA_L + l) * 2 + 0];
        const float ry = ref[((size_t)row * MSDA_L + l) * 2 + 1];
        const float* vbase = value + ((size_t)(n * MSDA_LQ + s0)) * 256 + h * 32 + lane;

        #pragma unroll
        for (int p = 0; p < MSDA_P; ++p) {
            const float ox = off[(l * 4 + p) * 2 + 0];
            const float oy = off[(l * 4 + p) * 2 + 1];
            const float lx = fminf(fmaxf(rx + ox, 0.0f), 1.0f);
            const float ly = fminf(fmaxf(ry + oy, 0.0f), 1.0f);
            const float x = lx * (float)ww - 0.5f;
            const float y = ly * (float)hh - 0.5f;
            const int x0 = (int)floorf(x);
            const int y0 = (int)floorf(y);
            const int x0i = min(max(x0, 0), ww - 1);
            const int x1i = min(max(x0 + 1, 0), ww - 1);
            const int y0i = min(max(y0, 0), hh - 1);
            const int y1i = min(max(y0 + 1, 0), hh - 1);
            const float x0f = (float)x0i, x1f = (float)x1i;
            const float y0f = (float)y0i, y1f = (float)y1i;
            const float wa = (x1f - x) * (y1f - y);
            const float wb = (x1f - x) * (y - y0f);
            const float wc = (x - x0f) * (y1f - y);
            const float wd = (x - x0f) * (y - y0f);
            const float va = vbase[(size_t)(y0i * ww + x0i) * 256];
            const float vb = vbase[(size_t)(y1i * ww + x0i) * 256];
            const float vc = vbase[(size_t)(y0i * ww + x1i) * 256];
            const float vd = vbase[(size_t)(y1i * ww + x1i) * 256];
            acc += (e[l * 4 + p] * inv) * (wa * va + wb * vb + wc * vc + wd * vd);
        }
    }
    inter[(size_t)row * 256 + h * 32 + lane] = acc;
}

// ---------------------------------------------------------------------------
extern "C" void kernel_launch(void* const* d_in, const int* in_sizes, int n_in,
                              void* d_out, int out_size, void* d_ws, size_t ws_size,
                              hipStream_t stream) {
    const float* query  = (const float*)d_in[0];
    const float* refpts = (const float*)d_in[1];
    const float* inpf   = (const float*)d_in[2];
    const int*   shapes = (const int*)d_in[3];
    const int*   starts = (const int*)d_in[4];
    const float* w_off  = (const float*)d_in[5];
    const float* b_off  = (const float*)d_in[6];
    const float* w_attn = (const float*)d_in[7];
    const float* b_attn = (const float*)d_in[8];
    const float* w_val  = (const float*)d_in[9];
    const float* b_val  = (const float*)d_in[10];
    const float* w_out  = (const float*)d_in[11];
    const float* b_out  = (const float*)d_in[12];
    float* out = (float*)d_out;

    char* ws = (char*)d_ws;
    const size_t szC = (size_t)MSDA_M * 256 * sizeof(float);   // 22.3 MB
    const size_t szA = (size_t)MSDA_M * 128 * sizeof(float);   // 11.1 MB
    float* v_val  = (float*)(ws);
    float* v_off  = (float*)(ws + szC);
    float* v_attn = (float*)(ws + 2 * szC);
    float* v_int  = (float*)(ws + 2 * szC + szA);

    const dim3 blk(256);
    const int mBlocks = MSDA_M / 128;   // 170 exactly

    // value = input_flatten @ w_val + b_val
    gemm_f32_wmma<<<dim3(mBlocks, 4), blk, 0, stream>>>(inpf, w_val, b_val, v_val, 256);
    // off = query @ w_off + b_off
    gemm_f32_wmma<<<dim3(mBlocks, 4), blk, 0, stream>>>(query, w_off, b_off, v_off, 256);
    // attn logits = query @ w_attn + b_attn
    gemm_f32_wmma<<<dim3(mBlocks, 2), blk, 0, stream>>>(query, w_attn, b_attn, v_attn, 128);
    // softmax + deformable bilinear sampling -> inter
    msda_sample<<<dim3((MSDA_N * MSDA_LQ * MSDA_H) / 8), blk, 0, stream>>>(
        refpts, v_off, v_attn, v_val, v_int, shapes, starts);
    // out = inter @ w_out + b_out
    gemm_f32_wmma<<<dim3(mBlocks, 4), blk, 0, stream>>>(v_int, w_out, b_out, out, 256);
}